// GATLayer_80513456931225
// MI455X (gfx1250) — compile-verified
//
#include <hip/hip_runtime.h>
#include <hip/hip_bf16.h>
#include <stdint.h>

// Problem constants (match reference)
#define N_NODES 100000
#define N_EDGES 3200000
#define IN_DIM  128
#define OUT_DIM 64

typedef float v2f __attribute__((ext_vector_type(2)));
typedef float v8f __attribute__((ext_vector_type(8)));

// ---------------------------------------------------------------------------
// Kernel 1: h = x @ W using V_WMMA_F32_16X16X4_F32 (native fp32 matrix path).
// 256 threads = 8 wave32; each wave computes a 16-row x 64-col tile of h.
// K=128 -> 32 WMMA steps of K=4. W staged in LDS (32 KB of the 320 KB WGP LDS).
//
// VGPR layouts per CDNA5 ISA 7.12.2:
//   A (16x4 f32): lanes 0-15 hold M=lane, K={0,1}; lanes 16-31 hold M=lane-16,
//                 K={2,3}  -> per-lane v2f = {x[m][k+kh], x[m][k+kh+1]}
//   B (4x16 f32): vgpr v: lanes 0-15 row K=v, lanes 16-31 row K=v+2, N=lane&15
//   C/D (16x16):  vgpr r: lanes 0-15 M=r, lanes 16-31 M=r+8, N=lane&15
// ---------------------------------------------------------------------------
__global__ void __launch_bounds__(256)
gat_gemm_xw_kernel(const float* __restrict__ x, const float* __restrict__ W,
                   float* __restrict__ h) {
  __shared__ float Wl[IN_DIM * OUT_DIM]; // 32 KB
  for (int i = threadIdx.x; i < IN_DIM * OUT_DIM; i += 256) Wl[i] = W[i];
  __syncthreads();

  const int lane  = threadIdx.x & 31;
  const int wid   = threadIdx.x >> 5;
  const int l     = lane & 15;
  const int khalf = (lane >> 4) << 1;   // 0 or 2 (K pair handled by this half)
  const int row0  = blockIdx.x * 128 + wid * 16;

  // Clamp A-fragment row for the tail tile so EXEC stays all-1s through WMMA.
  int arow = row0 + l;
  if (arow >= N_NODES) arow = N_NODES - 1;
  const float* __restrict__ xr = x + (size_t)arow * IN_DIM;

  v8f acc0 = {}, acc1 = {}, acc2 = {}, acc3 = {};
  for (int k = 0; k < IN_DIM; k += 4) {
    const int kk = k + khalf;
    v2f a;
    a.x = xr[kk];
    a.y = xr[kk + 1];
    const float* __restrict__ w0 = &Wl[kk * OUT_DIM + l];
    const float* __restrict__ w1 = &Wl[(kk + 1) * OUT_DIM + l];
    v2f b0, b1, b2, b3;
    b0.x = w0[0];  b0.y = w1[0];
    b1.x = w0[16]; b1.y = w1[16];
    b2.x = w0[32]; b2.y = w1[32];
    b3.x = w0[48]; b3.y = w1[48];
    acc0 = __builtin_amdgcn_wmma_f32_16x16x4_f32(false, a, false, b0, (short)0, acc0, false, false);
    acc1 = __builtin_amdgcn_wmma_f32_16x16x4_f32(false, a, false, b1, (short)0, acc1, false, false);
    acc2 = __builtin_amdgcn_wmma_f32_16x16x4_f32(false, a, false, b2, (short)0, acc2, false, false);
    acc3 = __builtin_amdgcn_wmma_f32_16x16x4_f32(false, a, false, b3, (short)0, acc3, false, false);
  }

  const int mhalf = (lane >> 4) << 3;   // 0 or 8
#pragma unroll
  for (int r = 0; r < 8; ++r) {
    const int gr = row0 + mhalf + r;
    if (gr < N_NODES) {
      float* __restrict__ hrow = h + (size_t)gr * OUT_DIM + l;
      hrow[0]  = acc0[r];
      hrow[16] = acc1[r];
      hrow[32] = acc2[r];
      hrow[48] = acc3[r];
    }
  }
}

// ---------------------------------------------------------------------------
// Kernel 2: per-node attention logits hs = h.a_src, ht = h.a_tgt.
// h was just written -> resident in 192MB L2; pure L2-bandwidth pass.
// ---------------------------------------------------------------------------
__global__ void __launch_bounds__(256)
gat_logits_kernel(const float* __restrict__ h, const float* __restrict__ a_src,
                  const float* __restrict__ a_tgt, float* __restrict__ hs,
                  float* __restrict__ ht) {
  const int i = blockIdx.x * 256 + threadIdx.x;
  if (i >= N_NODES) return;
  const float4* __restrict__ hr  = (const float4*)(h + (size_t)i * OUT_DIM);
  const float4* __restrict__ as4 = (const float4*)a_src;
  const float4* __restrict__ at4 = (const float4*)a_tgt;
  float s = 0.f, t = 0.f;
#pragma unroll
  for (int d = 0; d < OUT_DIM / 4; ++d) {
    const float4 v = hr[d], a = as4[d], b = at4[d];
    s += v.x * a.x + v.y * a.y + v.z * a.z + v.w * a.w;
    t += v.x * b.x + v.y * b.y + v.z * b.z + v.w * b.w;
  }
  hs[i] = s;
  ht[i] = t;
}

// ---------------------------------------------------------------------------
// Init: zero d_out and node accumulators (harness poisons them to 0xAA).
// nodemax holds order-preserving uint keys; init = key(-inf) = 0x007FFFFF.
// ---------------------------------------------------------------------------
__global__ void __launch_bounds__(256)
gat_init_kernel(float* __restrict__ out, unsigned* __restrict__ nodemax,
                float* __restrict__ nodesum) {
  const int i = blockIdx.x * 256 + threadIdx.x;
  if (i < N_NODES * OUT_DIM) out[i] = 0.f;
  if (i < N_NODES) {
    nodemax[i] = 0x007FFFFFu;  // order-key of -inf
    nodesum[i] = 0.f;
  }
}

// Monotone float <-> uint32 order key (total order for non-NaN floats).
__device__ __forceinline__ unsigned f32_key(float v) {
  unsigned b = __float_as_uint(v);
  return (b & 0x80000000u) ? ~b : (b | 0x80000000u);
}
__device__ __forceinline__ float f32_unkey(unsigned k) {
  return (k & 0x80000000u) ? __uint_as_float(k ^ 0x80000000u)
                           : __uint_as_float(~k);
}

// ---------------------------------------------------------------------------
// Kernel 3: edge logit + segment-max over target nodes (atomicMax on keys).
// ---------------------------------------------------------------------------
__global__ void __launch_bounds__(256)
gat_edge_max_kernel(const long long* __restrict__ ei,
                    const float* __restrict__ ew,
                    const float* __restrict__ hs, const float* __restrict__ ht,
                    float* __restrict__ eval, unsigned* __restrict__ nodemax) {
  const int e = blockIdx.x * 256 + threadIdx.x;
  if (e >= N_EDGES) return;
  if (e + 16384 < N_EDGES) __builtin_prefetch(&ei[e + 16384], 0, 0);
  const int s = (int)ei[e];
  const int t = (int)ei[(size_t)N_EDGES + e];
  float v = hs[s] + ht[t];
  v = (v > 0.f) ? v : 0.2f * v;   // leaky_relu, slope 0.2
  v *= ew[e];
  eval[e] = v;
  atomicMax(&nodemax[t], f32_key(v));
}

// ---------------------------------------------------------------------------
// Kernel 4: e_exp = exp(e - max[tgt]); segment-sum via f32 atomics (L2-res.)
// ---------------------------------------------------------------------------
__global__ void __launch_bounds__(256)
gat_edge_exp_kernel(const long long* __restrict__ ei,
                    float* __restrict__ eval,
                    const unsigned* __restrict__ nodemax,
                    float* __restrict__ nodesum) {
  const int e = blockIdx.x * 256 + threadIdx.x;
  if (e >= N_EDGES) return;
  const int t = (int)ei[(size_t)N_EDGES + e];
  const float m  = f32_unkey(nodemax[t]);
  const float ex = __expf(eval[e] - m);
  eval[e] = ex;               // overwrite logit with exp in place
  atomicAdd(&nodesum[t], ex);
}

// ---------------------------------------------------------------------------
// Kernel 5: alpha = e_exp / (sum[tgt]+1e-10); out[tgt] += alpha * h[src].
// One wave per edge, 2 dims per lane (float2 gather of h[src] from L2,
// 64 f32 atomic adds per edge into L2-resident out).
// ---------------------------------------------------------------------------
__global__ void __launch_bounds__(256)
gat_edge_scatter_kernel(const long long* __restrict__ ei,
                        const float* __restrict__ eval,
                        const float* __restrict__ nodesum,
                        const float* __restrict__ h, float* __restrict__ out) {
  const int e    = blockIdx.x * 8 + (threadIdx.x >> 5);
  const int lane = threadIdx.x & 31;
  if (e >= N_EDGES) return;
  const int s = (int)ei[e];
  const int t = (int)ei[(size_t)N_EDGES + e];
  const float alpha = eval[e] / (nodesum[t] + 1e-10f);
  const float2 v = ((const float2*)(h + (size_t)s * OUT_DIM))[lane];
  float* __restrict__ o = out + (size_t)t * OUT_DIM + lane * 2;
  atomicAdd(o + 0, v.x * alpha);
  atomicAdd(o + 1, v.y * alpha);
}

// ---------------------------------------------------------------------------
// Launch
//   d_in: 0=x(f32 N*128) 1=edge_index(i64 2*E) 2=edge_weight(f32 E)
//         3=W(f32 128*64) 4=a_src(f32 64) 5=a_tgt(f32 64)
//   d_out: f32 N*64
//   d_ws layout (bytes): h[25.6M] hs[0.4M] ht[0.4M] eval[12.8M]
//                        nodemax[0.4M] nodesum[0.4M]  (~40 MB total)
// ---------------------------------------------------------------------------
extern "C" void kernel_launch(void* const* d_in, const int* in_sizes, int n_in,
                              void* d_out, int out_size, void* d_ws, size_t ws_size,
                              hipStream_t stream) {
  const float*     x     = (const float*)d_in[0];
  const long long* ei    = (const long long*)d_in[1];
  const float*     ew    = (const float*)d_in[2];
  const float*     W     = (const float*)d_in[3];
  const float*     a_src = (const float*)d_in[4];
  const float*     a_tgt = (const float*)d_in[5];
  float*           out   = (float*)d_out;

  char* base = (char*)d_ws;
  float*    h       = (float*)base;
  float*    hs      = (float*)(base + (size_t)N_NODES * OUT_DIM * 4);
  float*    ht      = hs + N_NODES;
  float*    eval    = ht + N_NODES;
  unsigned* nodemax = (unsigned*)(eval + N_EDGES);
  float*    nodesum = (float*)(nodemax + N_NODES);

  // init out / node accumulators (covers N*OUT_DIM elements)
  gat_init_kernel<<<(N_NODES * OUT_DIM + 255) / 256, 256, 0, stream>>>(out, nodemax, nodesum);

  // h = x @ W  (WMMA fp32)
  gat_gemm_xw_kernel<<<(N_NODES + 127) / 128, 256, 0, stream>>>(x, W, h);

  // hs, ht
  gat_logits_kernel<<<(N_NODES + 255) / 256, 256, 0, stream>>>(h, a_src, a_tgt, hs, ht);

  // segment softmax over incoming edges
  gat_edge_max_kernel<<<(N_EDGES + 255) / 256, 256, 0, stream>>>(ei, ew, hs, ht, eval, nodemax);
  gat_edge_exp_kernel<<<(N_EDGES + 255) / 256, 256, 0, stream>>>(ei, eval, nodemax, nodesum);

  // weighted scatter-add aggregation (one wave per edge)
  gat_edge_scatter_kernel<<<(N_EDGES + 7) / 8, 256, 0, stream>>>(ei, eval, nodesum, h, out);
}